// MergePooling_9552007266577
// MI455X (gfx1250) — compile-verified
//
#include <hip/hip_runtime.h>
#include <math.h>

typedef __attribute__((ext_vector_type(2))) float v2f;
typedef __attribute__((ext_vector_type(8))) float v8f;

#define C_DIM 128
#define TWO_C 256

// ---------------------------------------------------------------------------
// Kernel 1: segment mean+max pooling over sorted batch indices.
// One block per graph; 256 threads = 8 row-groups x 32 lanes, float4 per lane.
// Writes merged[g, 0:128] = mean, merged[g, 128:256] = max into workspace.
// ---------------------------------------------------------------------------
__global__ __launch_bounds__(256) void pool_kernel(const float* __restrict__ x,
                                                   const int* __restrict__ batch,
                                                   float* __restrict__ merged,
                                                   int n_nodes) {
    const int g = blockIdx.x;

    __shared__ int s_bounds[2];
    if (threadIdx.x == 0) {
        // lower_bound(batch, g)
        int lo = 0, hi = n_nodes;
        while (lo < hi) { int mid = (lo + hi) >> 1; if (batch[mid] < g) lo = mid + 1; else hi = mid; }
        s_bounds[0] = lo;
        // lower_bound(batch, g+1), starting from previous result
        hi = n_nodes;
        while (lo < hi) { int mid = (lo + hi) >> 1; if (batch[mid] < g + 1) lo = mid + 1; else hi = mid; }
        s_bounds[1] = lo;
    }
    __syncthreads();
    const int start = s_bounds[0];
    const int end   = s_bounds[1];
    const int cnt   = end - start;

    const int lane4    = (threadIdx.x & 31) << 2;  // channel group: 4 channels per lane
    const int row_sub  = threadIdx.x >> 5;         // 0..7

    float4 sum = make_float4(0.f, 0.f, 0.f, 0.f);
    float4 mx  = make_float4(-INFINITY, -INFINITY, -INFINITY, -INFINITY);

    for (int i = start + row_sub; i < end; i += 8) {
        const float4 v = *(const float4*)(x + (size_t)i * C_DIM + lane4);
        sum.x += v.x; sum.y += v.y; sum.z += v.z; sum.w += v.w;
        mx.x = fmaxf(mx.x, v.x); mx.y = fmaxf(mx.y, v.y);
        mx.z = fmaxf(mx.z, v.z); mx.w = fmaxf(mx.w, v.w);
    }

    __shared__ float4 s_sum[256];
    __shared__ float4 s_max[256];
    s_sum[threadIdx.x] = sum;
    s_max[threadIdx.x] = mx;
    __syncthreads();

    if (threadIdx.x < 32) {
        float4 ts = s_sum[threadIdx.x];
        float4 tm = s_max[threadIdx.x];
        #pragma unroll
        for (int r = 1; r < 8; ++r) {
            const float4 os = s_sum[r * 32 + threadIdx.x];
            const float4 om = s_max[r * 32 + threadIdx.x];
            ts.x += os.x; ts.y += os.y; ts.z += os.z; ts.w += os.w;
            tm.x = fmaxf(tm.x, om.x); tm.y = fmaxf(tm.y, om.y);
            tm.z = fmaxf(tm.z, om.z); tm.w = fmaxf(tm.w, om.w);
        }
        float4 mean, mxo;
        if (cnt > 0) {
            const float inv = 1.0f / (float)cnt;
            mean = make_float4(ts.x * inv, ts.y * inv, ts.z * inv, ts.w * inv);
            mxo  = tm;
        } else {
            mean = make_float4(0.f, 0.f, 0.f, 0.f);
            mxo  = make_float4(0.f, 0.f, 0.f, 0.f);
        }
        float* mrow = merged + (size_t)g * TWO_C;
        *(float4*)(mrow + lane4)         = mean;   // mean in [0,128)
        *(float4*)(mrow + C_DIM + lane4) = mxo;    // max  in [128,256)
    }
}

// ---------------------------------------------------------------------------
// Kernel 2: alpha = sigmoid(merged @ W + b); out = alpha*mean + (1-alpha)*max
// merged: [G, 256], W: [256, 128], out: [G, 128].
// One 16x16 output tile per wave via V_WMMA_F32_16X16X4_F32, K-loop of 64.
// Block = 8 waves covering the 128 output columns; grid = G/16 blocks.
//
// Operand layouts (ISA 7.12.2, 32-bit, wave32):
//   A 16x4 : lanes 0-15 -> M, lane-half selects K pair {0,1}/{2,3} across 2 VGPRs
//   B 4x16 : lanes 0-15 -> N, lane-half selects K pair across 2 VGPRs
//   C/D    : VGPR r, lane L -> M = r + 8*(L>=16), N = L & 15
// ---------------------------------------------------------------------------
__global__ __launch_bounds__(256) void gemm_gate_kernel(const float* __restrict__ merged,
                                                        const float* __restrict__ W,
                                                        const float* __restrict__ bias,
                                                        float* __restrict__ out) {
    const int g0   = blockIdx.x * 16;        // 16 graphs (M tile)
    const int wave = threadIdx.x >> 5;       // 0..7 -> N tile
    const int lane = threadIdx.x & 31;
    const int n0   = wave * 16;
    const int half = lane >> 4;              // K-pair selector within WMMA step
    const int mn   = lane & 15;              // M index for A, N index for B

    const float* arow = merged + (size_t)(g0 + mn) * TWO_C;  // A row for this lane

    v8f acc = {};
    #pragma unroll 4
    for (int k0 = 0; k0 < TWO_C; k0 += 4) {
        const int k = k0 + half * 2;
        v2f a, b;
        a.x = arow[k];
        a.y = arow[k + 1];
        b.x = W[(size_t)k * C_DIM + n0 + mn];
        b.y = W[(size_t)(k + 1) * C_DIM + n0 + mn];
        acc = __builtin_amdgcn_wmma_f32_16x16x4_f32(
            /*neg_a=*/false, a, /*neg_b=*/false, b,
            /*c_mod=*/(short)0, acc, /*reuse_a=*/false, /*reuse_b=*/false);
    }

    // Epilogue: bias + sigmoid gate, blend mean/max, store.
    const int n  = n0 + mn;                  // output column for this lane
    const float bn = bias[n];
    #pragma unroll
    for (int r = 0; r < 8; ++r) {
        const int m  = r + half * 8;
        const int gg = g0 + m;
        const float pre   = acc[r] + bn;
        const float alpha = 1.0f / (1.0f + __expf(-pre));
        const float* mrow = merged + (size_t)gg * TWO_C;
        const float meanv = mrow[n];
        const float maxv  = mrow[C_DIM + n];
        out[(size_t)gg * C_DIM + n] = alpha * meanv + (1.0f - alpha) * maxv;
    }
}

// ---------------------------------------------------------------------------
extern "C" void kernel_launch(void* const* d_in, const int* in_sizes, int n_in,
                              void* d_out, int out_size, void* d_ws, size_t ws_size,
                              hipStream_t stream) {
    const float* x     = (const float*)d_in[0];
    const int*   batch = (const int*)d_in[1];
    const float* W     = (const float*)d_in[2];
    const float* b     = (const float*)d_in[3];
    float*       out   = (float*)d_out;
    float*       merged = (float*)d_ws;      // [n_graphs, 256] fp32 = 8 MB

    const int n_nodes  = in_sizes[0] / C_DIM;   // 2,000,000
    const int n_graphs = out_size / C_DIM;      // 8192

    pool_kernel<<<n_graphs, 256, 0, stream>>>(x, batch, merged, n_nodes);
    gemm_gate_kernel<<<n_graphs / 16, 256, 0, stream>>>(merged, W, b, out);
}